// EncoderModel_1589137899700
// MI455X (gfx1250) — compile-verified
//
#include <hip/hip_runtime.h>

typedef __attribute__((ext_vector_type(2))) float v2f;
typedef __attribute__((ext_vector_type(8))) float v8f;

#define SIZE 32
#define VOX (SIZE * SIZE * SIZE) /* 32768 */
#define NPTS 1024

__device__ __forceinline__ int clampi(int x, int hi) {
  return min(max(x, 0), hi);
}

// ---------------------------------------------------------------------------
// Kernel 1: build_images — per-cluster min/max reduce, then trilinear
// scatter-add into a 32^3 LDS image (ds_add_f32 atomics), then stream out.
// ---------------------------------------------------------------------------
__global__ __launch_bounds__(256) void k_build(const float* __restrict__ voxels,
                                               const float* __restrict__ values,
                                               float* __restrict__ x0_out) {
  __shared__ float img[VOX];           // 128 KB (CDNA5: 320 KB LDS per WGP)
  __shared__ float smin[3 * 256];
  __shared__ float smax[3 * 256];
  __shared__ float sparam[4];          // off[3], grid_step

  const int c = blockIdx.x;
  const int tid = threadIdx.x;
  const float* V = voxels + (size_t)c * NPTS * 3;
  const float* W = values + (size_t)c * NPTS;

  float mn[3] = {1e30f, 1e30f, 1e30f};
  float mx[3] = {-1e30f, -1e30f, -1e30f};
  for (int i = tid; i < NPTS; i += 256) {
    for (int a = 0; a < 3; a++) {
      float v = V[i * 3 + a];
      mn[a] = fminf(mn[a], v);
      mx[a] = fmaxf(mx[a], v);
    }
  }
  for (int a = 0; a < 3; a++) { smin[a * 256 + tid] = mn[a]; smax[a * 256 + tid] = mx[a]; }
  __syncthreads();
  for (int s = 128; s > 0; s >>= 1) {
    if (tid < s) {
      for (int a = 0; a < 3; a++) {
        smin[a * 256 + tid] = fminf(smin[a * 256 + tid], smin[a * 256 + tid + s]);
        smax[a * 256 + tid] = fmaxf(smax[a * 256 + tid], smax[a * 256 + tid + s]);
      }
    }
    __syncthreads();
  }
  if (tid == 0) {
    float rng[3], maxr = -1e30f;
    for (int a = 0; a < 3; a++) { rng[a] = smax[a * 256] - smin[a * 256]; maxr = fmaxf(maxr, rng[a]); }
    maxr += 1.0f;
    float gs = 1.0f / maxr;
    for (int a = 0; a < 3; a++)
      sparam[a] = -(smin[a * 256] + rng[a] * 0.5f + 0.5f) * gs + 0.5f;
    sparam[3] = gs;
  }
  __syncthreads();
  for (int i = tid; i < VOX; i += 256) img[i] = 0.0f;
  __syncthreads();

  const float gs = sparam[3];
  const float inv32 = 1.0f / 32.0f;
  for (int i = tid; i < NPTS; i += 256) {
    float gm[3];
    int base[3];
    for (int a = 0; a < 3; a++) {
      gm[a] = V[i * 3 + a] * gs + sparam[a];
      base[a] = (int)floorf(gm[a] * 32.0f);
    }
    float val = W[i];
    for (int corner = 0; corner < 8; corner++) {
      int id3[3] = {base[0] + ((corner >> 2) & 1),
                    base[1] + ((corner >> 1) & 1),
                    base[2] + (corner & 1)};
      float wgt = val;
      int ok = 1;
      for (int a = 0; a < 3; a++) {
        int idx = id3[a];
        float lo = fmaxf(gm[a], idx * inv32);
        float hi = fminf(gm[a] + gs, (idx + 1) * inv32);
        wgt *= fmaxf((hi - lo) / gs, 0.0f);
        ok &= (unsigned)idx < 32u;
      }
      if (ok) atomicAdd(&img[(id3[0] * 32 + id3[1]) * 32 + id3[2]], wgt);
    }
  }
  __syncthreads();
  float* out = x0_out + (size_t)c * VOX;
  for (int i = tid; i < VOX; i += 256) out[i] = img[i];
}

// ---------------------------------------------------------------------------
// Kernel 2: conv1 (1->2 ch, k=4, s=2, p=1, ReLU) as implicit GEMM via
// V_WMMA_F32_16X16X4_F32.  M=2(pad 16), K=64, N=4096 (tiles of 16).
// B gathers are unconditional (clamped address) + value select -> no exec
// masking; A fragments are tile-invariant registers from a zero-padded
// 16x64 weight matrix in LDS.
// ---------------------------------------------------------------------------
__device__ __forceinline__ float tapB1(const float* img, int od, int oh, int ow, int k) {
  int kd = k >> 4, kh = (k >> 2) & 3, kw = k & 3;
  int id = od * 2 + kd - 1, ih = oh * 2 + kh - 1, iw = ow * 2 + kw - 1;
  int ok = ((unsigned)id < 32u) & ((unsigned)ih < 32u) & ((unsigned)iw < 32u);
  float v = img[(clampi(id, 31) << 10) + (clampi(ih, 31) << 5) + clampi(iw, 31)];
  return ok ? v : 0.0f;
}

__global__ __launch_bounds__(256) void k_conv1(const float* __restrict__ x0,
                                               const float* __restrict__ w1,
                                               const float* __restrict__ b1,
                                               float* __restrict__ y1) {
  __shared__ float img[VOX];   // 128 KB input tile
  __shared__ float swp[1024];  // zero-padded weights [16][64]
  __shared__ float sb[2];
  const int c = blockIdx.x, tid = threadIdx.x;
  const float* in = x0 + (size_t)c * VOX;
  for (int i = tid; i < VOX; i += 256) img[i] = in[i];
  for (int i = tid; i < 1024; i += 256) swp[i] = (i < 128) ? w1[i] : 0.0f;
  if (tid < 2) sb[tid] = b1[tid];
  __syncthreads();

  const int wave = tid >> 5, lane = tid & 31;
  const int m = lane & 15, half = lane >> 4;
  float* out = y1 + (size_t)c * 8192;

  // Tile-invariant A fragments (kept in VGPRs).
  v2f av[16];
#pragma unroll
  for (int ch = 0; ch < 16; ch++) {
    int kb = ch * 4 + half * 2;
    av[ch].x = swp[m * 64 + kb];
    av[ch].y = swp[m * 64 + kb + 1];
  }

  for (int t = wave; t < 256; t += 8) {       // 32 tiles per wave, uniform
    int od = t >> 4, oh = t & 15;
    v8f acc = {};
#pragma unroll
    for (int ch = 0; ch < 16; ch++) {         // K = 64 in chunks of 4
      int kb = ch * 4 + half * 2;
      v2f b;
      b.x = tapB1(img, od, oh, m, kb);
      b.y = tapB1(img, od, oh, m, kb + 1);
      acc = __builtin_amdgcn_wmma_f32_16x16x4_f32(false, av[ch], false, b,
                                                  (short)0, acc, false, false);
    }
    if (half == 0) {                          // rows 0,1 = out channels
      int p = od * 256 + oh * 16 + m;
      out[p]        = fmaxf(acc[0] + sb[0], 0.0f);
      out[4096 + p] = fmaxf(acc[1] + sb[1], 0.0f);
    }
  }
}

// ---------------------------------------------------------------------------
// Kernel 3: conv2 (2->4 ch, k=4, s=2, p=1, ReLU) via WMMA. K=128, N=512.
// ---------------------------------------------------------------------------
__device__ __forceinline__ float tapB2(const float* img, int od, int oh, int ow, int k) {
  int ic = k >> 6, r = k & 63;
  int kd = r >> 4, kh = (r >> 2) & 3, kw = r & 3;
  int id = od * 2 + kd - 1, ih = oh * 2 + kh - 1, iw = ow * 2 + kw - 1;
  int ok = ((unsigned)id < 16u) & ((unsigned)ih < 16u) & ((unsigned)iw < 16u);
  float v = img[ic * 4096 + (clampi(id, 15) << 8) + (clampi(ih, 15) << 4) + clampi(iw, 15)];
  return ok ? v : 0.0f;
}

__global__ __launch_bounds__(256) void k_conv2(const float* __restrict__ y1,
                                               const float* __restrict__ w2,
                                               const float* __restrict__ b2,
                                               float* __restrict__ y2) {
  __shared__ float img[8192];  // 32 KB [2][16^3]
  __shared__ float swp[2048];  // zero-padded weights [16][128]
  __shared__ float sb[4];
  const int c = blockIdx.x, tid = threadIdx.x;
  const float* in = y1 + (size_t)c * 8192;
  for (int i = tid; i < 8192; i += 256) img[i] = in[i];
  for (int i = tid; i < 2048; i += 256) swp[i] = (i < 512) ? w2[i] : 0.0f;
  if (tid < 4) sb[tid] = b2[tid];
  __syncthreads();

  const int wave = tid >> 5, lane = tid & 31;
  const int m = lane & 15, half = lane >> 4;
  float* out = y2 + (size_t)c * 2048;

  v2f av[32];
#pragma unroll
  for (int ch = 0; ch < 32; ch++) {
    int kb = ch * 4 + half * 2;
    av[ch].x = swp[m * 128 + kb];
    av[ch].y = swp[m * 128 + kb + 1];
  }

  for (int t = wave; t < 32; t += 8) {        // 4 tiles per wave, uniform
    int p = t * 16 + m;
    int od = p >> 6, oh = (p >> 3) & 7, ow = p & 7;
    v8f acc = {};
#pragma unroll
    for (int ch = 0; ch < 32; ch++) {         // K = 128 in chunks of 4
      int kb = ch * 4 + half * 2;
      v2f b;
      b.x = tapB2(img, od, oh, ow, kb);
      b.y = tapB2(img, od, oh, ow, kb + 1);
      acc = __builtin_amdgcn_wmma_f32_16x16x4_f32(false, av[ch], false, b,
                                                  (short)0, acc, false, false);
    }
    if (half == 0) {
      for (int oc = 0; oc < 4; oc++)
        out[oc * 512 + p] = fmaxf(acc[oc] + sb[oc], 0.0f);
    }
  }
}

// ---------------------------------------------------------------------------
// Kernel 4: conv3 (4->8 ch, k=3, s=2, p=0, no ReLU) -> embedding [C,216]
// ---------------------------------------------------------------------------
__global__ __launch_bounds__(256) void k_conv3(const float* __restrict__ y2,
                                               const float* __restrict__ w3,
                                               const float* __restrict__ b3,
                                               float* __restrict__ emb) {
  const int c = blockIdx.x, tid = threadIdx.x;
  if (tid >= 216) return;
  const float* in = y2 + (size_t)c * 2048;
  int oc = tid / 27, p = tid % 27;
  int od = p / 9, oh = (p / 3) % 3, ow = p % 3;
  float acc = b3[oc];
  for (int ic = 0; ic < 4; ic++)
    for (int kd = 0; kd < 3; kd++)
      for (int kh = 0; kh < 3; kh++)
        for (int kw = 0; kw < 3; kw++) {
          int id = od * 2 + kd, ih = oh * 2 + kh, iw = ow * 2 + kw;
          acc += in[ic * 512 + (id << 6) + (ih << 3) + iw] *
                 w3[oc * 108 + ic * 27 + kd * 9 + kh * 3 + kw];
        }
  emb[(size_t)c * 216 + tid] = acc;
}

// ---------------------------------------------------------------------------
// Transposed conv helper: for output coord o (k=4, stride 2, pad), at most
// two (input, kernel) pairs per axis: kd = o + pad - 2*id in [0,4).
// ---------------------------------------------------------------------------
__device__ __forceinline__ int axis_pairs(int o, int pad, int din, int* ids, int* ks) {
  int cnt = 0;
  int t = o + pad;
  int id0 = t >> 1, k0 = t & 1;
  if (id0 >= 0 && id0 < din) { ids[cnt] = id0; ks[cnt] = k0; cnt++; }
  int id1 = id0 - 1;
  if (id1 >= 0 && id1 < din) { ids[cnt] = id1; ks[cnt] = k0 + 2; cnt++; }
  return cnt;
}

// Kernel 5: convT1 (8->4 ch, k=4, s=2, p=0, ReLU): [8,3^3] -> [4,8^3]
__global__ __launch_bounds__(256) void k_dconv1(const float* __restrict__ emb,
                                                const float* __restrict__ dw1,
                                                const float* __restrict__ db1,
                                                float* __restrict__ y) {
  __shared__ float sx[216];
  __shared__ float sw[2048];  // dw1 [8][4][64]
  __shared__ float sb[4];
  const int c = blockIdx.x, tid = threadIdx.x;
  const float* in = emb + (size_t)c * 216;
  if (tid < 216) sx[tid] = in[tid];
  for (int i = tid; i < 2048; i += 256) sw[i] = dw1[i];
  if (tid < 4) sb[tid] = db1[tid];
  __syncthreads();
  float* out = y + (size_t)c * 2048;
  for (int o = tid; o < 2048; o += 256) {
    int oc = o >> 9, p = o & 511;
    int od = p >> 6, oh = (p >> 3) & 7, ow = p & 7;
    int iD[2], kD[2], iH[2], kH[2], iW[2], kW[2];
    int nd = axis_pairs(od, 0, 3, iD, kD);
    int nh = axis_pairs(oh, 0, 3, iH, kH);
    int nw = axis_pairs(ow, 0, 3, iW, kW);
    float acc = sb[oc];
    for (int a = 0; a < nd; a++)
      for (int b = 0; b < nh; b++)
        for (int d = 0; d < nw; d++) {
          int koff = kD[a] * 16 + kH[b] * 4 + kW[d];
          int xoff = (iD[a] * 3 + iH[b]) * 3 + iW[d];
          for (int ic = 0; ic < 8; ic++)
            acc += sx[ic * 27 + xoff] * sw[ic * 256 + oc * 64 + koff];
        }
    out[o] = fmaxf(acc, 0.0f);
  }
}

// Kernel 6: convT2 (4->2 ch, k=4, s=2, p=1, ReLU): [4,8^3] -> [2,16^3]
__global__ __launch_bounds__(256) void k_dconv2(const float* __restrict__ x,
                                                const float* __restrict__ dw2,
                                                const float* __restrict__ db2,
                                                float* __restrict__ y) {
  __shared__ float sx[2048];
  __shared__ float sw[512];  // dw2 [4][2][64]
  __shared__ float sb[2];
  const int c = blockIdx.x, tid = threadIdx.x;
  const float* in = x + (size_t)c * 2048;
  for (int i = tid; i < 2048; i += 256) sx[i] = in[i];
  for (int i = tid; i < 512; i += 256) sw[i] = dw2[i];
  if (tid < 2) sb[tid] = db2[tid];
  __syncthreads();
  float* out = y + (size_t)c * 8192;
  for (int o = tid; o < 8192; o += 256) {
    int oc = o >> 12, p = o & 4095;
    int od = p >> 8, oh = (p >> 4) & 15, ow = p & 15;
    int iD[2], kD[2], iH[2], kH[2], iW[2], kW[2];
    int nd = axis_pairs(od, 1, 8, iD, kD);
    int nh = axis_pairs(oh, 1, 8, iH, kH);
    int nw = axis_pairs(ow, 1, 8, iW, kW);
    float acc = sb[oc];
    for (int a = 0; a < nd; a++)
      for (int b = 0; b < nh; b++)
        for (int d = 0; d < nw; d++) {
          int koff = kD[a] * 16 + kH[b] * 4 + kW[d];
          int xoff = (iD[a] << 6) + (iH[b] << 3) + iW[d];
          for (int ic = 0; ic < 4; ic++)
            acc += sx[ic * 512 + xoff] * sw[ic * 128 + oc * 64 + koff];
        }
    out[o] = fmaxf(acc, 0.0f);
  }
}

// Kernel 7: convT3 (2->1 ch, k=4, s=2, p=1, no ReLU): [2,16^3] -> [1,32^3]
__global__ __launch_bounds__(256) void k_dconv3(const float* __restrict__ x,
                                                const float* __restrict__ dw3,
                                                const float* __restrict__ db3,
                                                float* __restrict__ y) {
  __shared__ float sx[8192];  // 32 KB
  __shared__ float sw[128];   // dw3 [2][64]
  __shared__ float sb0;
  const int c = blockIdx.x, tid = threadIdx.x;
  const float* in = x + (size_t)c * 8192;
  for (int i = tid; i < 8192; i += 256) sx[i] = in[i];
  if (tid < 128) sw[tid] = dw3[tid];
  if (tid == 0) sb0 = db3[0];
  __syncthreads();
  float* out = y + (size_t)c * VOX;
  for (int o = tid; o < VOX; o += 256) {
    int od = o >> 10, oh = (o >> 5) & 31, ow = o & 31;
    int iD[2], kD[2], iH[2], kH[2], iW[2], kW[2];
    int nd = axis_pairs(od, 1, 16, iD, kD);
    int nh = axis_pairs(oh, 1, 16, iH, kH);
    int nw = axis_pairs(ow, 1, 16, iW, kW);
    float acc = sb0;
    for (int a = 0; a < nd; a++)
      for (int b = 0; b < nh; b++)
        for (int d = 0; d < nw; d++) {
          int koff = kD[a] * 16 + kH[b] * 4 + kW[d];
          int xoff = (iD[a] << 8) + (iH[b] << 4) + iW[d];
          acc += sx[xoff] * sw[koff] + sx[4096 + xoff] * sw[64 + koff];
        }
    out[o] = acc;
  }
}

// ---------------------------------------------------------------------------
extern "C" void kernel_launch(void* const* d_in, const int* in_sizes, int n_in,
                              void* d_out, int out_size, void* d_ws, size_t ws_size,
                              hipStream_t stream) {
  const float* voxels = (const float*)d_in[0];
  const float* values = (const float*)d_in[1];
  const float* w1 = (const float*)d_in[2];
  const float* b1 = (const float*)d_in[3];
  const float* w2 = (const float*)d_in[4];
  const float* b2 = (const float*)d_in[5];
  const float* w3 = (const float*)d_in[6];
  const float* b3 = (const float*)d_in[7];
  const float* dw1 = (const float*)d_in[8];
  const float* db1 = (const float*)d_in[9];
  const float* dw2 = (const float*)d_in[10];
  const float* db2 = (const float*)d_in[11];
  const float* dw3 = (const float*)d_in[12];
  const float* db3 = (const float*)d_in[13];

  const int C = in_sizes[0] / (NPTS * 3);  // 1024

  float* out = (float*)d_out;
  float* Y   = out;                              // [C,1,32,32,32]
  float* EMB = out + (size_t)C * VOX;            // [C,216]
  float* X0  = EMB + (size_t)C * 216;            // [C,1,32,32,32]

  float* bufA = (float*)d_ws;                    // C*2*4096 floats (33.5 MB)
  float* bufB = bufA + (size_t)C * 8192;         // C*4*512  floats (8.4 MB)

  dim3 grid(C), block(256);
  k_build <<<grid, block, 0, stream>>>(voxels, values, X0);
  k_conv1 <<<grid, block, 0, stream>>>(X0, w1, b1, bufA);
  k_conv2 <<<grid, block, 0, stream>>>(bufA, w2, b2, bufB);
  k_conv3 <<<grid, block, 0, stream>>>(bufB, w3, b3, EMB);
  k_dconv1<<<grid, block, 0, stream>>>(EMB, dw1, db1, bufB);  // reuse bufB
  k_dconv2<<<grid, block, 0, stream>>>(bufB, dw2, db2, bufA); // reuse bufA
  k_dconv3<<<grid, block, 0, stream>>>(bufA, dw3, db3, Y);
}